// SparseMultiHeadedAttention_53506702573974
// MI455X (gfx1250) — compile-verified
//
#include <hip/hip_runtime.h>

typedef __attribute__((ext_vector_type(2))) float v2f;
typedef __attribute__((ext_vector_type(8))) float v8f;

#define EMBED    1024
#define HEADS    16
#define HEAD_DIM 64
#define BATCH    4
#define SEQ      1024
#define NH       (BATCH * HEADS)   // 64 head-batches

__device__ __forceinline__ v8f wmma_f32_16x16x4(v2f a, v2f b, v8f c) {
    // 8-arg pattern: (neg_a, A, neg_b, B, c_mod, C, reuse_a, reuse_b)
    return __builtin_amdgcn_wmma_f32_16x16x4_f32(false, a, false, b,
                                                 (short)0, c, false, false);
}

// Per-lane A/B fragment pointer for V_WMMA_F32_16X16X4_F32:
// lanes 0-15 hold K={0,1}, lanes 16-31 hold K={2,3}  ->  b64 load at +half*2.
__device__ __forceinline__ const v2f* frag_ptr(const float* base, int row,
                                               int pitch, int half) {
    return (const v2f*)(base + (size_t)row * pitch + half * 2);
}

// ---------------------------------------------------------------------------
// Kernel 1: Q/K projection.  C[4096 x 2048] = x[4096 x 1024] @ W[f,:]^T with
// f = 1024 + col (k-chunk cols 0..1023 and q-chunk cols 1024..2047 are
// contiguous rows of W).  One wave per 32x32 tile (2x2 WMMA accumulators);
// output scattered to head-major layout [n][s][d] in workspace.
// ---------------------------------------------------------------------------
__global__ __launch_bounds__(256) void qk_proj_kernel(
    const float* __restrict__ x, const float* __restrict__ W,
    float* __restrict__ kh, float* __restrict__ qh) {
    const int lane = threadIdx.x & 31;
    const int wave = threadIdx.x >> 5;
    const int tile = blockIdx.x * 8 + wave;       // 8192 waves total
    const int tM = tile >> 6;                     // 128 along M (32-rows)
    const int tN = tile & 63;                     // 64 along N (32-cols)
    const int half = lane >> 4;
    const int n16  = lane & 15;

    const v2f* __restrict__ a0p = frag_ptr(x, tM * 32 + n16,        EMBED, half);
    const v2f* __restrict__ a1p = frag_ptr(x, tM * 32 + 16 + n16,   EMBED, half);
    const v2f* __restrict__ b0p = frag_ptr(W, 1024 + tN * 32 + n16,      EMBED, half);
    const v2f* __restrict__ b1p = frag_ptr(W, 1024 + tN * 32 + 16 + n16, EMBED, half);

    v8f acc00 = {}, acc01 = {}, acc10 = {}, acc11 = {};
#pragma unroll 4
    for (int kk = 0; kk < EMBED / 4; ++kk) {      // K=4 per step, b64 per lane
        v2f a0 = a0p[kk * 2];
        v2f a1 = a1p[kk * 2];
        v2f b0 = b0p[kk * 2];
        v2f b1 = b1p[kk * 2];
        acc00 = wmma_f32_16x16x4(a0, b0, acc00);
        acc01 = wmma_f32_16x16x4(a0, b1, acc01);
        acc10 = wmma_f32_16x16x4(a1, b0, acc10);
        acc11 = wmma_f32_16x16x4(a1, b1, acc11);
    }

    // Scatter the four 16x16 sub-tiles to head-major [n=b*16+h][s][d]
    v8f accs[2][2] = {{acc00, acc01}, {acc10, acc11}};
#pragma unroll
    for (int ni = 0; ni < 2; ++ni) {
        const int col = tN * 32 + ni * 16 + n16;  // 0..2047
        float* __restrict__ dst = (col < 1024) ? kh : qh;
        const int cc = col & 1023;
        const int h = cc >> 6, dd = cc & 63;
#pragma unroll
        for (int mi = 0; mi < 2; ++mi) {
#pragma unroll
            for (int r = 0; r < 8; ++r) {
                int m = tM * 32 + mi * 16 + half * 8 + r;   // [0,4096)
                int bidx = m >> 10, s = m & 1023;
                dst[(((size_t)(bidx * HEADS + h)) * SEQ + s) * HEAD_DIM + dd] =
                    accs[mi][ni][r];
            }
        }
    }
}

// ---------------------------------------------------------------------------
// Kernel 2: per-head logits  L[n] = Q[n] @ K[n]^T   ([1024x64]x[64x1024]).
// One wave per 32x32 tile (2x2 accumulators), 16 K-steps over d=64.
// ---------------------------------------------------------------------------
__global__ __launch_bounds__(256) void logits_kernel(
    const float* __restrict__ qh, const float* __restrict__ kh,
    float* __restrict__ out) {
    const int lane = threadIdx.x & 31;
    const int wave = threadIdx.x >> 5;
    const int tile = blockIdx.x * 8 + wave;       // 65536 waves total
    const int n   = tile >> 10;                   // head-batch 0..63
    const int rem = tile & 1023;
    const int tq  = rem >> 5;                     // 32 q-tiles of 32 rows
    const int tk  = rem & 31;                     // 32 k-tiles of 32 cols
    const int half = lane >> 4;
    const int n16  = lane & 15;

    const float* qn = qh + (size_t)n * SEQ * HEAD_DIM;
    const float* kn = kh + (size_t)n * SEQ * HEAD_DIM;
    const v2f* __restrict__ a0p = frag_ptr(qn, tq * 32 + n16,      HEAD_DIM, half);
    const v2f* __restrict__ a1p = frag_ptr(qn, tq * 32 + 16 + n16, HEAD_DIM, half);
    const v2f* __restrict__ b0p = frag_ptr(kn, tk * 32 + n16,      HEAD_DIM, half);
    const v2f* __restrict__ b1p = frag_ptr(kn, tk * 32 + 16 + n16, HEAD_DIM, half);

    v8f acc00 = {}, acc01 = {}, acc10 = {}, acc11 = {};
#pragma unroll
    for (int kk = 0; kk < HEAD_DIM / 4; ++kk) {
        v2f a0 = a0p[kk * 2];
        v2f a1 = a1p[kk * 2];
        v2f b0 = b0p[kk * 2];
        v2f b1 = b1p[kk * 2];
        acc00 = wmma_f32_16x16x4(a0, b0, acc00);
        acc01 = wmma_f32_16x16x4(a0, b1, acc01);
        acc10 = wmma_f32_16x16x4(a1, b0, acc10);
        acc11 = wmma_f32_16x16x4(a1, b1, acc11);
    }

    float* __restrict__ orow = out + (size_t)n * SEQ * SEQ;
    v8f accs[2][2] = {{acc00, acc01}, {acc10, acc11}};
#pragma unroll
    for (int mi = 0; mi < 2; ++mi) {
#pragma unroll
        for (int ni = 0; ni < 2; ++ni) {
            const int col = tk * 32 + ni * 16 + n16;
#pragma unroll
            for (int r = 0; r < 8; ++r) {
                int m = tq * 32 + mi * 16 + half * 8 + r;
                orow[(size_t)m * SEQ + col] = accs[mi][ni][r];
            }
        }
    }
}

// ---------------------------------------------------------------------------
// Kernel 3: sparsemax over each row of 1024, in place on d_out.
// One 256-thread block per row: bitonic sort (descending) in LDS, cumsum via
// 4-per-thread + Hillis-Steele scan, support size k, tau, relu.
// ---------------------------------------------------------------------------
__global__ __launch_bounds__(256) void sparsemax_kernel(float* __restrict__ out) {
    __shared__ float zs[1024];
    __shared__ float cs[1024];
    __shared__ float part[256];

    const int tid = threadIdx.x;
    float* row = out + (size_t)blockIdx.x * 1024;
    float4 v = ((float4*)row)[tid];

    // row max
    float m = fmaxf(fmaxf(v.x, v.y), fmaxf(v.z, v.w));
    part[tid] = m;
    __syncthreads();
    for (int off = 128; off > 0; off >>= 1) {
        if (tid < off) part[tid] = fmaxf(part[tid], part[tid + off]);
        __syncthreads();
    }
    float rmax = part[0];
    __syncthreads();

    v.x -= rmax; v.y -= rmax; v.z -= rmax; v.w -= rmax;
    const int base = tid * 4;
    zs[base + 0] = v.x; zs[base + 1] = v.y; zs[base + 2] = v.z; zs[base + 3] = v.w;
    __syncthreads();

    // bitonic sort, descending
    for (int ksz = 2; ksz <= 1024; ksz <<= 1) {
        for (int j = ksz >> 1; j > 0; j >>= 1) {
#pragma unroll
            for (int t = 0; t < 4; ++t) {
                int i = tid + t * 256;
                int ixj = i ^ j;
                if (ixj > i) {
                    float a = zs[i], b = zs[ixj];
                    bool desc = ((i & ksz) == 0);
                    if (desc ? (a < b) : (a > b)) { zs[i] = b; zs[ixj] = a; }
                }
            }
            __syncthreads();
        }
    }

    // inclusive cumsum of sorted values
    float l0 = zs[base], l1 = zs[base + 1], l2 = zs[base + 2], l3 = zs[base + 3];
    float s0 = l0, s1 = s0 + l1, s2 = s1 + l2, s3 = s2 + l3;
    part[tid] = s3;
    __syncthreads();
    for (int off = 1; off < 256; off <<= 1) {
        float pv = part[tid];
        float add = (tid >= off) ? part[tid - off] : 0.0f;
        __syncthreads();
        part[tid] = pv + add;
        __syncthreads();
    }
    float offs = part[tid] - s3;
    cs[base + 0] = offs + s0; cs[base + 1] = offs + s1;
    cs[base + 2] = offs + s2; cs[base + 3] = offs + s3;

    // support size k = #{r : 1 + r*zs[r-1] > cumsum[r-1]} (prefix property)
    int cnt = 0;
#pragma unroll
    for (int t = 0; t < 4; ++t) {
        float r = (float)(base + t + 1);
        if (1.0f + r * zs[base + t] > cs[base + t]) ++cnt;
    }
    __syncthreads();
    part[tid] = (float)cnt;
    __syncthreads();
    for (int off = 128; off > 0; off >>= 1) {
        if (tid < off) part[tid] += part[tid + off];
        __syncthreads();
    }
    int ksupp = (int)part[0];
    __syncthreads();
    float tau = (cs[ksupp - 1] - 1.0f) / (float)ksupp;

    float4 o;
    o.x = fmaxf(0.0f, v.x - tau);
    o.y = fmaxf(0.0f, v.y - tau);
    o.z = fmaxf(0.0f, v.z - tau);
    o.w = fmaxf(0.0f, v.w - tau);
    ((float4*)row)[tid] = o;
}

// ---------------------------------------------------------------------------
extern "C" void kernel_launch(void* const* d_in, const int* in_sizes, int n_in,
                              void* d_out, int out_size, void* d_ws, size_t ws_size,
                              hipStream_t stream) {
    (void)in_sizes; (void)n_in; (void)out_size; (void)ws_size;
    const float* x = (const float*)d_in[0];         // [4,1024,1024] fp32
    const float* W = (const float*)d_in[1];         // [3072,1024]   fp32
    float* out = (float*)d_out;                     // [64,1024,1024] fp32

    float* kh = (float*)d_ws;                       // [64][1024][64] (16 MB)
    float* qh = kh + (size_t)NH * SEQ * HEAD_DIM;   // [64][1024][64] (16 MB)

    qk_proj_kernel<<<1024, 256, 0, stream>>>(x, W, kh, qh);
    logits_kernel<<<8192, 256, 0, stream>>>(qh, kh, out);
    sparsemax_kernel<<<65536, 256, 0, stream>>>(out);
}